// FourGrad_47545287967269
// MI455X (gfx1250) — compile-verified
//
#include <hip/hip_runtime.h>
#include <stdint.h>

// FourGrad fused: 4 depthwise 3x3 filters -> Laplacian -> 4-way variance.
// Store-bound: 128 MiB read + 640 MiB written => ~35us floor @ 23.3 TB/s.
// CDNA5 features used: Tensor Data Mover (tensor_load_to_lds) with double
// buffering over TENSORcnt, b128 LDS/global accesses, nontemporal output
// stores. WMMA deliberately unused (fp32 16x16x4 Toeplitz form <15% useful
// MACs; kernel is at the HBM roofline regardless).

#define HH 256
#define WW 256
#define TH 8                 // output rows per tile
#define INROWS (TH + 4)      // input rows incl. halo-2 (two chained 3x3)
#define SROWS  (TH + 2)      // stage-A rows incl. halo-1
#define NT 2                 // tiles per block, double-buffered TDM

typedef uint32_t u32;
typedef __attribute__((ext_vector_type(4))) u32   v4u;
typedef __attribute__((ext_vector_type(8))) int   v8i;
typedef __attribute__((ext_vector_type(4))) int   v4i;
typedef __attribute__((ext_vector_type(4))) float v4f;

// Issue one TDM 2-D tile DMA: nrows x 256 fp32, global (row-major, stride 256
// elements) -> LDS at lds_byte. D# layout per CDNA5 ISA ch.10.8.
__device__ __forceinline__ void tdm_load_tile(const float* gsrc, u32 lds_byte,
                                              int nrows) {
  const uint64_t gaddr = (uint64_t)(uintptr_t)gsrc;

  v4u g0;                                     // group 0 (128b)
  g0.x = 1u;                                  // count=1, user descriptor
  g0.y = lds_byte;                            // lds_addr
  g0.z = (u32)gaddr;                          // global_addr[31:0]
  g0.w = ((u32)(gaddr >> 32) & 0x01FFFFFFu) | (2u << 30);  // [56:32] | type=2

  v8i g1;                                     // group 1 (256b)
  g1[0] = (int)(2u << 16);                    // wg_mask=0, data_size=4B
  g1[1] = (int)(((u32)WW & 0xFFFFu) << 16);   // tensor_dim0 lo16 @ [31:16]
  g1[2] = (int)((((u32)WW) >> 16) | (((u32)nrows & 0xFFFFu) << 16));
  g1[3] = (int)((((u32)nrows) >> 16) | (((u32)WW & 0xFFFFu) << 16)); // tile_dim0
  g1[4] = (int)((u32)nrows & 0xFFFFu);        // tile_dim1 (tile_dim2=0)
  g1[5] = (int)WW;                            // tensor_dim0_stride lo32
  g1[6] = 0;
  g1[7] = 0;

  v4i gz4 = {0, 0, 0, 0};                     // 2-D tensor: groups 2/3 unused
#if defined(__clang_major__) && (__clang_major__ >= 23)
  v8i gz8 = {0, 0, 0, 0, 0, 0, 0, 0};
  __builtin_amdgcn_tensor_load_to_lds(g0, g1, gz4, gz4, gz8, 0);
#else
  __builtin_amdgcn_tensor_load_to_lds(g0, g1, gz4, gz4, 0);
#endif
}

// Fused stencil on one staged tile. Each thread owns a 4-column strip
// (tx -> cols 4tx..4tx+3) so LDS and global traffic is b128.
__device__ __forceinline__ void compute_tile(
    const float (*in_t)[WW], float (*sA)[SROWS][WW],
    float* __restrict__ o0, float* __restrict__ o1, float* __restrict__ o2,
    float* __restrict__ o3, float* __restrict__ ov,
    size_t gbase, int tx, int ty)
{
  const int c0 = tx * 4;

  // ---- Stage A: sobel_x/y + diagonals on SROWS rows --------------------
  for (int sr = ty; sr < SROWS; sr += 4) {
    v4f r0v = *(const v4f*)&in_t[sr][c0];
    v4f r1v = *(const v4f*)&in_t[sr + 1][c0];
    v4f r2v = *(const v4f*)&in_t[sr + 2][c0];
    float l0 = (c0 > 0) ? in_t[sr][c0 - 1]     : 0.0f;
    float l1 = (c0 > 0) ? in_t[sr + 1][c0 - 1] : 0.0f;
    float l2 = (c0 > 0) ? in_t[sr + 2][c0 - 1] : 0.0f;
    float e0 = (c0 + 4 < WW) ? in_t[sr][c0 + 4]     : 0.0f;
    float e1 = (c0 + 4 < WW) ? in_t[sr + 1][c0 + 4] : 0.0f;
    float e2 = (c0 + 4 < WW) ? in_t[sr + 2][c0 + 4] : 0.0f;

    v4f sx, sy, d1, d2;
#pragma unroll
    for (int j = 0; j < 4; ++j) {
      float a  = (j == 0) ? l0 : r0v[j - 1];
      float b  = r0v[j];
      float c  = (j == 3) ? e0 : r0v[j + 1];
      float d  = (j == 0) ? l1 : r1v[j - 1];
      float f  = (j == 3) ? e1 : r1v[j + 1];
      float g  = (j == 0) ? l2 : r2v[j - 1];
      float h  = r2v[j];
      float ii = (j == 3) ? e2 : r2v[j + 1];
      sx[j] = (a - c) + (g - ii) + 2.0f * (d - f);     // sobel_x
      sy[j] = (a + c - g - ii) + 2.0f * (b - h);       // sobel_y
      d1[j] = (b - d) + (f - h) + 2.0f * (c - g);      // diag1
      d2[j] = (b - f) + (d - h) + 2.0f * (a - ii);     // diag2
    }
    *(v4f*)&sA[0][sr][c0] = sx;
    *(v4f*)&sA[1][sr][c0] = sy;
    *(v4f*)&sA[2][sr][c0] = d1;
    *(v4f*)&sA[3][sr][c0] = d2;
  }
  __syncthreads();

  // ---- Stage B: Laplacian + unbiased variance, b128 NT stores ----------
#pragma unroll
  for (int rr = 0; rr < TH / 4; ++rr) {
    const int r = ty + rr * 4;                 // rows ty and ty+4
    v4f v[4];
#pragma unroll
    for (int k = 0; k < 4; ++k) {
      v4f top = *(const v4f*)&sA[k][r][c0];
      v4f mid = *(const v4f*)&sA[k][r + 1][c0];
      v4f bot = *(const v4f*)&sA[k][r + 2][c0];
      float lm = (c0 > 0)      ? sA[k][r + 1][c0 - 1] : 0.0f;
      float rm = (c0 + 4 < WW) ? sA[k][r + 1][c0 + 4] : 0.0f;
#pragma unroll
      for (int j = 0; j < 4; ++j) {
        float lf = (j == 0) ? lm : mid[j - 1];
        float rt = (j == 3) ? rm : mid[j + 1];
        v[k][j] = (top[j] + bot[j] + lf + rt) - 4.0f * mid[j];
      }
    }
    v4f mean = 0.25f * (v[0] + v[1] + v[2] + v[3]);
    v4f q0 = v[0] - mean, q1 = v[1] - mean, q2 = v[2] - mean, q3 = v[3] - mean;
    v4f var = (q0 * q0 + q1 * q1 + q2 * q2 + q3 * q3) * (1.0f / 3.0f);

    const size_t idx = gbase + (size_t)r * WW;
    __builtin_nontemporal_store(v[0], (v4f*)&o0[idx]);
    __builtin_nontemporal_store(v[1], (v4f*)&o1[idx]);
    __builtin_nontemporal_store(v[2], (v4f*)&o2[idx]);
    __builtin_nontemporal_store(v[3], (v4f*)&o3[idx]);
    __builtin_nontemporal_store(var,  (v4f*)&ov[idx]);
  }
}

__global__ __launch_bounds__(256)
void fourgrad_fused_kernel(const float* __restrict__ x,
                           float* __restrict__ out,
                           int planes)
{
  __shared__ float in_t[NT][INROWS][WW];       // 24 KB, double-buffered tiles
  __shared__ float sA[4][SROWS][WW];           // 40 KB first-conv results

  const int tid = threadIdx.x;
  const int tx  = tid & 63;                    // 4-col strip index
  const int ty  = tid >> 6;                    // 0..3
  const int groups = (HH / TH) / NT;           // 16 strip-pairs per plane
  const int plane  = blockIdx.x / groups;
  const int grp    = blockIdx.x % groups;

  const float* xp = x + (size_t)plane * HH * WW;
  const size_t N  = (size_t)planes * HH * WW;
  float* o0 = out;
  float* o1 = out + N;
  float* o2 = out + 2 * N;
  float* o3 = out + 3 * N;
  float* ov = out + 4 * N;

  // Per-tile staging geometry (clamp halo rows to the plane) + zero-fill of
  // out-of-plane halo rows (only first/last strips of a plane do work here).
  int lo[NT], nrows[NT], skip[NT];
#pragma unroll
  for (int t = 0; t < NT; ++t) {
    const int r0    = (grp * NT + t) * TH;
    const int gr_lo = r0 - 2;
    const int l  = (gr_lo < 0) ? 0 : gr_lo;
    const int h0 = r0 + TH + 1;
    const int h  = (h0 > HH - 1) ? (HH - 1) : h0;
    lo[t] = l;  nrows[t] = h - l + 1;  skip[t] = l - gr_lo;
    const int tail = INROWS - (skip[t] + nrows[t]);
    float* bf = &in_t[t][0][0];
    for (int i = tid; i < skip[t] * WW; i += 256) bf[i] = 0.0f;
    for (int i = tid; i < tail * WW; i += 256)
      bf[(skip[t] + nrows[t]) * WW + i] = 0.0f;
  }

  // Issue both tile DMAs from wave 0, then wait for tile 0 only: tile 1's
  // DMA overlaps tile 0's compute (TENSORcnt is in-order per wave).
  if (tid < 32) {
#pragma unroll
    for (int t = 0; t < NT; ++t)
      tdm_load_tile(xp + (size_t)lo[t] * WW,
                    (u32)(uintptr_t)&in_t[t][skip[t]][0], nrows[t]);
    __builtin_amdgcn_s_wait_tensorcnt(NT - 1);
  }
  __syncthreads();

#pragma unroll
  for (int t = 0; t < NT; ++t) {
    const int r0 = (grp * NT + t) * TH;
    const size_t gbase =
        (size_t)plane * HH * WW + (size_t)r0 * WW + (size_t)(tx * 4);
    compute_tile(in_t[t], sA, o0, o1, o2, o3, ov, gbase, tx, ty);
    __syncthreads();                           // sA reusable for next tile
    if (t + 1 < NT) {
      if (tid < 32) __builtin_amdgcn_s_wait_tensorcnt(0);
      __syncthreads();
    }
  }
}

extern "C" void kernel_launch(void* const* d_in, const int* in_sizes, int n_in,
                              void* d_out, int out_size, void* d_ws, size_t ws_size,
                              hipStream_t stream) {
  (void)in_sizes; (void)n_in; (void)out_size; (void)d_ws; (void)ws_size;
  const float* x = (const float*)d_in[0];
  float* out = (float*)d_out;
  const int planes = 8 * 64;                            // B*C
  const int grid = planes * ((HH / TH) / NT);           // 8192 blocks
  fourgrad_fused_kernel<<<grid, 256, 0, stream>>>(x, out, planes);
}